// Attention_71545565217156
// MI455X (gfx1250) — compile-verified
//
#include <hip/hip_runtime.h>
#include <math.h>

#define DEC 128
#define ENC 128

typedef float v2f __attribute__((ext_vector_type(2)));
typedef float v4f __attribute__((ext_vector_type(4)));
typedef float v8f __attribute__((ext_vector_type(8)));

// ---------------------------------------------------------------------------
// float atomic max via signed/unsigned integer atomics (works for all signs,
// smax initialized to -inf).
// ---------------------------------------------------------------------------
__device__ __forceinline__ void atomicMaxFloat(float* addr, float val) {
    if (val >= 0.0f) {
        atomicMax((int*)addr, __float_as_int(val));
    } else {
        atomicMin((unsigned int*)addr, __float_as_uint(val));
    }
}

// ---------------------------------------------------------------------------
// Pass 0: zero output, init smax=-inf, denom=0 (harness poisons buffers).
// ---------------------------------------------------------------------------
__global__ void k_init(float* __restrict__ out, int out_n,
                       float* __restrict__ smax, float* __restrict__ denom,
                       int batch) {
    int i = blockIdx.x * blockDim.x + threadIdx.x;
    if (i < out_n)  out[i] = 0.0f;
    if (i < batch) { smax[i] = -INFINITY; denom[i] = 0.0f; }
}

// ---------------------------------------------------------------------------
// Pass 1: qW = query @ W  via V_WMMA_F32_16X16X4_F32.
// One wave per 16x16 output tile, K=128 swept in steps of 4.
// A (16x4 f32): lanes 0-15 -> M=lane, K={0,1}; lanes 16-31 -> M=lane-16, K={2,3}
// B (4x16 f32): lanes 0-15 -> N=lane, K={0,1}; lanes 16-31 -> N=lane-16, K={2,3}
// C/D (16x16):  vgpr v, lanes 0-15 -> M=v, N=lane; lanes 16-31 -> M=v+8.
// ---------------------------------------------------------------------------
__global__ void k_qw_wmma(const float* __restrict__ query,
                          const float* __restrict__ W,
                          float* __restrict__ qW, int batch) {
    const int wave = blockIdx.x * (blockDim.x >> 5) + (threadIdx.x >> 5);
    const int lane = threadIdx.x & 31;
    const int tilesN = ENC / 16;
    const int tm = wave / tilesN;
    const int tn = wave % tilesN;
    if (tm * 16 >= batch) return;            // wave-uniform guard (EXEC stays all-1s)

    const int half = lane >> 4;              // 0: lanes 0-15, 1: lanes 16-31
    const int l15  = lane & 15;
    const int arow = tm * 16 + l15;          // M index held by this lane
    const int bcol = tn * 16 + l15;          // N index held by this lane

    v8f c = {};
    for (int k = 0; k < DEC; k += 4) {
        const int kb = k + half * 2;         // this lane's K base (0/1 or 2/3 of the step)
        v2f a = *(const v2f*)(query + arow * DEC + kb);   // K consecutive -> float2
        v2f b;
        b.x = W[(kb)     * ENC + bcol];
        b.y = W[(kb + 1) * ENC + bcol];
        c = __builtin_amdgcn_wmma_f32_16x16x4_f32(
                /*neg_a=*/false, a, /*neg_b=*/false, b,
                /*c_mod=*/(short)0, c, /*reuse_a=*/false, /*reuse_b=*/false);
    }
#pragma unroll
    for (int v = 0; v < 8; ++v) {
        const int row = tm * 16 + v + half * 8;
        qW[row * ENC + bcol] = c[v];
    }
}

// ---------------------------------------------------------------------------
// Pass 2: s[n] = <qW[index[n]], values[n]> / sqrt(128); segment max via atomics.
// One wave32 per node: lane loads float4 -> 512B coalesced line per node.
// values is streamed (512MB, > L2) -> non-temporal; qW (4MB) stays in L2.
// ---------------------------------------------------------------------------
__global__ void k_scores(const float* __restrict__ values,
                         const float* __restrict__ qW,
                         const int* __restrict__ index,
                         float* __restrict__ s, float* __restrict__ smax,
                         int nodes) {
    const int node = blockIdx.x * (blockDim.x >> 5) + (threadIdx.x >> 5);
    const int lane = threadIdx.x & 31;
    if (node >= nodes) return;

    const int idx = index[node];
    v4f v = __builtin_nontemporal_load((const v4f*)(values + (size_t)node * ENC) + lane);
    v4f q = ((const v4f*)(qW + (size_t)idx * ENC))[lane];

    float sum = v.x * q.x + v.y * q.y + v.z * q.z + v.w * q.w;
#pragma unroll
    for (int off = 16; off > 0; off >>= 1)
        sum += __shfl_xor(sum, off, 32);
    sum *= 0.08838834764831845f;             // 1/sqrt(128)

    if (lane == 0) {
        s[node] = sum;
        atomicMaxFloat(smax + idx, sum);
    }
}

// ---------------------------------------------------------------------------
// Pass 3: e[n] = exp(s[n] - smax[idx]); denom[idx] += e (stable softmax).
// ---------------------------------------------------------------------------
__global__ void k_exp(const int* __restrict__ index, float* __restrict__ s,
                      const float* __restrict__ smax, float* __restrict__ denom,
                      int nodes) {
    const int i = blockIdx.x * blockDim.x + threadIdx.x;
    if (i >= nodes) return;
    const int idx = index[i];
    const float e = expf(s[i] - smax[idx]);
    s[i] = e;                                 // reuse buffer for e
    atomicAdd(denom + idx, e);
}

// ---------------------------------------------------------------------------
// Pass 4: out[idx] += (e[n]/denom[idx]) * values[n]. Second streaming pass
// over values; atomic f32 adds land in the 4MB (L2-resident) output.
// ---------------------------------------------------------------------------
__global__ void k_pool(const float* __restrict__ values,
                       const int* __restrict__ index,
                       const float* __restrict__ e,
                       const float* __restrict__ denom,
                       float* __restrict__ out, int nodes) {
    const int node = blockIdx.x * (blockDim.x >> 5) + (threadIdx.x >> 5);
    const int lane = threadIdx.x & 31;
    if (node >= nodes) return;

    const int idx = index[node];
    const float w = e[node] / denom[idx];
    v4f v = __builtin_nontemporal_load((const v4f*)(values + (size_t)node * ENC) + lane);

    float* o = out + (size_t)idx * ENC + lane * 4;
    atomicAdd(o + 0, w * v.x);
    atomicAdd(o + 1, w * v.y);
    atomicAdd(o + 2, w * v.z);
    atomicAdd(o + 3, w * v.w);
}

// ---------------------------------------------------------------------------
extern "C" void kernel_launch(void* const* d_in, const int* in_sizes, int n_in,
                              void* d_out, int out_size, void* d_ws, size_t ws_size,
                              hipStream_t stream) {
    const float* query  = (const float*)d_in[0];   // [B, DEC]
    const float* values = (const float*)d_in[1];   // [N, ENC]
    const int*   index  = (const int*)d_in[2];     // [N]
    const float* W      = (const float*)d_in[3];   // [DEC, ENC]

    const int batch = in_sizes[0] / DEC;
    const int nodes = in_sizes[2];
    float* out = (float*)d_out;

    // workspace layout: qW [batch*ENC] | s/e [nodes] | smax [batch] | denom [batch]
    char* ws = (char*)d_ws;
    float* qW    = (float*)ws;
    float* s     = (float*)(ws + (size_t)batch * ENC * sizeof(float));
    float* smax  = (float*)((char*)s + (size_t)nodes * sizeof(float));
    float* denom = smax + batch;

    const int initN = (out_size > batch) ? out_size : batch;
    k_init<<<(initN + 255) / 256, 256, 0, stream>>>(out, out_size, smax, denom, batch);

    const int tiles = ((batch + 15) / 16) * (ENC / 16);
    k_qw_wmma<<<(tiles + 3) / 4, 128, 0, stream>>>(query, W, qW, batch);

    const int waveBlocks = (nodes + 7) / 8;        // 8 waves (256 threads) per block
    k_scores<<<waveBlocks, 256, 0, stream>>>(values, qW, index, s, smax, nodes);

    k_exp<<<(nodes + 255) / 256, 256, 0, stream>>>(index, s, smax, denom, nodes);

    k_pool<<<waveBlocks, 256, 0, stream>>>(values, index, s, denom, out, nodes);
}